// Transformer_74680891343402
// MI455X (gfx1250) — compile-verified
//
#include <hip/hip_runtime.h>
#include <hip/hip_bf16.h>

#define N_NODES 50000
#define D_MODEL 256
#define H 8
#define D_K 32
#define HK (H * D_K)   // 256

typedef __attribute__((ext_vector_type(2))) float v2f;
typedef __attribute__((ext_vector_type(8))) float v8f;

// ---------------------------------------------------------------------------
// Phase 1: fused QKV projection.  One wave computes one 16x16 output tile for
// q, k AND v (A-tile loads amortized 3x).  Full fp32 via V_WMMA_F32_16X16X4_F32.
// 50000 rows = 3125 M-tiles, 256 cols = 16 N-tiles -> exactly 50000 waves.
// ---------------------------------------------------------------------------
__global__ __launch_bounds__(256) void qkv_gemm_wmma(
    const float* __restrict__ x,
    const float* __restrict__ Wq, const float* __restrict__ Wk,
    const float* __restrict__ Wv,
    float* __restrict__ q, float* __restrict__ k, float* __restrict__ v)
{
    const int wave = (blockIdx.x * blockDim.x + threadIdx.x) >> 5;  // wave32
    const int lane = threadIdx.x & 31;
    const int tm = wave >> 4;        // M-tile (0..3124)
    const int tn = wave & 15;        // N-tile (0..15)

    // A layout (16x4 f32): lane L holds M=L&15; VGPR0 K=klo, VGPR1 K=klo+1,
    // klo = 2*(L>>4).  B layout (4x16): lane L holds N=L&15 at same K pattern.
    const int m   = lane & 15;
    const int klo = (lane >> 4) << 1;
    const int col = tn * 16 + (lane & 15);
    const float* __restrict__ xrow = x + (size_t)(tm * 16 + m) * D_MODEL;

    v8f cq = {}, ck = {}, cv = {};
    for (int k0 = 0; k0 < D_MODEL; k0 += 4) {
        v2f a;
        a.x = xrow[k0 + klo];
        a.y = xrow[k0 + klo + 1];
        const int b0 = (k0 + klo) * HK + col;
        v2f bq, bk, bv;
        bq.x = Wq[b0]; bq.y = Wq[b0 + HK];
        bk.x = Wk[b0]; bk.y = Wk[b0 + HK];
        bv.x = Wv[b0]; bv.y = Wv[b0 + HK];
        cq = __builtin_amdgcn_wmma_f32_16x16x4_f32(false, a, false, bq, (short)0, cq, false, false);
        ck = __builtin_amdgcn_wmma_f32_16x16x4_f32(false, a, false, bk, (short)0, ck, false, false);
        cv = __builtin_amdgcn_wmma_f32_16x16x4_f32(false, a, false, bv, (short)0, cv, false, false);
    }

    // C/D layout: VGPR r -> row M = r + 8*(lane>=16), col N = lane&15.
    const int mhi = (lane >> 4) << 3;
    for (int r = 0; r < 8; ++r) {
        const size_t o = (size_t)(tm * 16 + mhi + r) * HK + col;
        q[o] = cq[r];
        k[o] = ck[r];
        v[o] = cv[r];
    }
}

// ---------------------------------------------------------------------------
// Phase 2: one wave per edge.  Each lane owns 8 contiguous floats (32B) of the
// 256-float node row; head h = lane>>2 (4 lanes per 32-dim head).  Dot products
// reduced with __shfl_xor inside the 4-lane head group; exp(clip(.)); scatter
// v*score with f32 atomics (q/k/v/wv: 4 x 51.2MB = 204.8MB, more than 192MB).
// ---------------------------------------------------------------------------
__global__ __launch_bounds__(256) void edge_attn(
    const float* __restrict__ q, const float* __restrict__ k,
    const float* __restrict__ v,
    const int* __restrict__ src, const int* __restrict__ dst,
    float* __restrict__ wv, float* __restrict__ z, int n_edges)
{
    const int edge = (blockIdx.x * blockDim.x + threadIdx.x) >> 5;
    if (edge >= n_edges) return;
    const int lane = threadIdx.x & 31;
    const int s = src[edge];
    const int d = dst[edge];

    const float4* kp = (const float4*)(k + (size_t)s * HK) + lane * 2;
    const float4* qp = (const float4*)(q + (size_t)d * HK) + lane * 2;
    const float4 k0 = kp[0], k1 = kp[1];
    const float4 q0 = qp[0], q1 = qp[1];

    float part = k0.x * q0.x + k0.y * q0.y + k0.z * q0.z + k0.w * q0.w
               + k1.x * q1.x + k1.y * q1.y + k1.z * q1.z + k1.w * q1.w;
    // reduce over the 4-lane head group (all 4 lanes end with the full dot)
    part += __shfl_xor(part, 1, 32);
    part += __shfl_xor(part, 2, 32);

    float sc = part * 0.1767766952966369f;          // 1/sqrt(32)
    sc = fminf(5.0f, fmaxf(-5.0f, sc));
    sc = __expf(sc);

    const float4* vp = (const float4*)(v + (size_t)s * HK) + lane * 2;
    const float4 v0 = vp[0], v1 = vp[1];
    float* out = wv + (size_t)d * HK + lane * 8;
    atomicAdd(out + 0, v0.x * sc);
    atomicAdd(out + 1, v0.y * sc);
    atomicAdd(out + 2, v0.z * sc);
    atomicAdd(out + 3, v0.w * sc);
    atomicAdd(out + 4, v1.x * sc);
    atomicAdd(out + 5, v1.y * sc);
    atomicAdd(out + 6, v1.z * sc);
    atomicAdd(out + 7, v1.w * sc);
    if ((lane & 3) == 0)
        atomicAdd(z + (size_t)d * H + (lane >> 2), sc);
}

// ---------------------------------------------------------------------------
// Phase 3: in-place normalization of the accumulated wv (= d_out) by z.
// ---------------------------------------------------------------------------
__global__ __launch_bounds__(256) void normalize_out(
    float* __restrict__ out, const float* __restrict__ z)
{
    const size_t i = (size_t)blockIdx.x * 256 + threadIdx.x;  // < 50000*256
    const int n = (int)(i >> 8);
    const int c = (int)(i & 255);
    out[i] = out[i] / (z[n * H + (c >> 5)] + 1e-9f);
}

extern "C" void kernel_launch(void* const* d_in, const int* in_sizes, int n_in,
                              void* d_out, int out_size, void* d_ws, size_t ws_size,
                              hipStream_t stream)
{
    const float* x  = (const float*)d_in[0];
    const float* Wq = (const float*)d_in[1];
    const float* Wk = (const float*)d_in[2];
    const float* Wv = (const float*)d_in[3];
    const int* src  = (const int*)d_in[4];
    const int* dst  = (const int*)d_in[5];
    const int n_edges = in_sizes[4];

    float* out = (float*)d_out;                       // doubles as wv accumulator
    float* q = (float*)d_ws;
    float* k = q + (size_t)N_NODES * HK;
    float* v = k + (size_t)N_NODES * HK;
    float* z = v + (size_t)N_NODES * HK;              // N_NODES * H floats

    // zero the accumulators (d_out/d_ws are poisoned by the harness)
    hipMemsetAsync(out, 0, (size_t)N_NODES * HK * sizeof(float), stream);
    hipMemsetAsync(z,   0, (size_t)N_NODES * H  * sizeof(float), stream);

    // Phase 1: 50000 tiles, 8 waves per 256-thread block -> 6250 blocks.
    qkv_gemm_wmma<<<(N_NODES * 16) / 8 / 16, 256, 0, stream>>>(x, Wq, Wk, Wv, q, k, v);

    // Phase 2: one wave per edge, 8 waves per block.
    const int eblocks = (n_edges + 7) / 8;
    edge_attn<<<eblocks, 256, 0, stream>>>(q, k, v, src, dst, out, z, n_edges);

    // Phase 3: one thread per output element.
    normalize_out<<<N_NODES, 256, 0, stream>>>(out, z);
}